// GNNQNetwork_5334349382302
// MI455X (gfx1250) — compile-verified
//
#include <hip/hip_runtime.h>
#include <hip/hip_bf16.h>
#include <stdint.h>

// ---------------------------------------------------------------------------
// 2-layer GCN for MI455X (gfx1250, wave32).
//   deg/dinv -> [XW1 via V_WMMA_F32_16X16X4_F32] -> edge scatter (f32 atomics)
//   -> bias+relu -> [HW2 via WMMA] -> edge scatter -> bias+relu (in d_out)
// ---------------------------------------------------------------------------

typedef __attribute__((ext_vector_type(2))) float v2f;
typedef __attribute__((ext_vector_type(8))) float v8f;

#define FDIM 96   // IN_DIM == HID == 96

// ---------------- degree / normalization ----------------------------------

__global__ void k_fill1(float* __restrict__ p, int n) {
    int i = blockIdx.x * blockDim.x + threadIdx.x;
    if (i < n) p[i] = 1.0f;   // self-loop contributes 1 to every degree
}

__global__ void k_deg_edges(const long long* __restrict__ ei,
                            float* __restrict__ deg, int nE) {
    int e = blockIdx.x * blockDim.x + threadIdx.x;
    if (e < nE) {
        int d = (int)ei[nE + e];          // edge_index[1][e] = target
        atomicAdd(&deg[d], 1.0f);
    }
}

__global__ void k_rsqrt_inplace(float* __restrict__ p, int n) {
    int i = blockIdx.x * blockDim.x + threadIdx.x;
    if (i < n) p[i] = __frsqrt_rn(p[i]); // deg >= 1 always (self loops)
}

// ---------------- utility ---------------------------------------------------

__global__ void k_zero(float* __restrict__ p, int n) {
    int i = blockIdx.x * blockDim.x + threadIdx.x;
    if (i < n) p[i] = 0.0f;
}

__global__ void k_bias_relu(const float* __restrict__ agg,
                            const float* __restrict__ bias,
                            float* __restrict__ out, int total) {
    int i = blockIdx.x * blockDim.x + threadIdx.x;
    if (i < total) {
        float v = agg[i] + bias[i % FDIM];
        out[i] = v > 0.0f ? v : 0.0f;
    }
}

// ---------------- GEMM: Y[M x 96] = X[M x 96] @ W[96 x 96] -----------------
// One wave32 per 16x16 output tile; K swept 4 at a time with
// V_WMMA_F32_16X16X4_F32 (exact f32 semantics, matches reference).
//
// Fragment layouts (ISA 7.12.2, wave32):
//   A 16x4 f32 : lanes 0-15 row M=l hold K = k0+0 (v0), k0+1 (v1)
//                lanes 16-31 row M=l hold K = k0+2 (v0), k0+3 (v1)
//   B 4x16 f32 : lanes 0-15 col N=l hold K = k0+0 (v0), k0+1 (v1)
//                lanes 16-31 col N=l hold K = k0+2 (v0), k0+3 (v1)
//   C/D 16x16  : VGPR r -> (M=r, N=lane) for lanes 0-15; (M=r+8, N=lane-16)
__global__ void k_gemm_wmma_f32(const float* __restrict__ X,
                                const float* __restrict__ W,
                                float* __restrict__ Y,
                                int nTiles, int tilesPerRow) {
    int wave = (int)((blockIdx.x * blockDim.x + threadIdx.x) >> 5);
    if (wave >= nTiles) return;              // wave-uniform: EXEC all-1 below
    int lane = threadIdx.x & 31;
    int half = lane >> 4;                    // 0: lanes 0-15, 1: lanes 16-31
    int l    = lane & 15;
    int koff = half * 2;

    int tm = wave / tilesPerRow;
    int tn = wave - tm * tilesPerRow;
    int row0 = tm * 16;
    int col0 = tn * 16;

    const float* xrow = X + (size_t)(row0 + l) * FDIM;

    v8f acc = {};
#pragma unroll
    for (int k0 = 0; k0 < FDIM; k0 += 4) {
        v2f a, b;
        // 8B-aligned pair load (row*96 + even offset)
        const float2 av = *(const float2*)(xrow + k0 + koff);
        a.x = av.x;  a.y = av.y;
        b.x = W[(size_t)(k0 + koff)     * FDIM + col0 + l];
        b.y = W[(size_t)(k0 + koff + 1) * FDIM + col0 + l];
        acc = __builtin_amdgcn_wmma_f32_16x16x4_f32(
            /*neg_a=*/false, a, /*neg_b=*/false, b,
            /*c_mod=*/(short)0, acc, /*reuse_a=*/false, /*reuse_b=*/false);
    }

#pragma unroll
    for (int r = 0; r < 8; ++r) {
        Y[(size_t)(row0 + half * 8 + r) * FDIM + col0 + l] = acc[r];
    }
}

// ---------------- message gather + scatter-add -----------------------------
// One wave32 per (edge | self-loop). Each lane does 3 non-returning f32
// atomics (96 cols / 32 lanes) -> GLOBAL_ATOMIC_ADD_F32 at the L2 atomic
// units. Source rows (19.2 MB matrix) live in the 192 MB L2.
__global__ void k_scatter(const long long* __restrict__ ei,
                          const float* __restrict__ dinv,
                          const float* __restrict__ feat,
                          float* __restrict__ agg,
                          int nEdges, int nNodes) {
    long long gtid = (long long)blockIdx.x * blockDim.x + threadIdx.x;
    int item = (int)(gtid >> 5);             // wave-uniform
    int lane = threadIdx.x & 31;
    int nItems = nEdges + nNodes;
    if (item >= nItems) return;

    int s, d;
    float w;
    if (item < nEdges) {
        s = (int)ei[item];                   // source j
        d = (int)ei[nEdges + item];          // target i
        w = dinv[s] * dinv[d];
    } else {                                 // self loop
        s = d = item - nEdges;
        float dv = dinv[s];
        w = dv * dv;
    }

    const float* __restrict__ row  = feat + (size_t)s * FDIM;
    float*       __restrict__ orow = agg  + (size_t)d * FDIM;
#pragma unroll
    for (int c = 0; c < 3; ++c) {
        int col = lane + 32 * c;
        atomicAdd(&orow[col], row[col] * w); // no-return f32 atomic
    }
}

// ---------------------------------------------------------------------------

extern "C" void kernel_launch(void* const* d_in, const int* in_sizes, int n_in,
                              void* d_out, int out_size, void* d_ws, size_t ws_size,
                              hipStream_t stream) {
    const float*     x  = (const float*)d_in[0];
    const long long* ei = (const long long*)d_in[1];   // int64 [2, E]
    const float*     W1 = (const float*)d_in[2];
    const float*     b1 = (const float*)d_in[3];
    const float*     W2 = (const float*)d_in[4];
    const float*     b2 = (const float*)d_in[5];
    float*           out = (float*)d_out;

    const int N = in_sizes[0] / FDIM;   // 50000 (divisible by 16)
    const int E = in_sizes[1] / 2;      // 800000
    const int total = N * FDIM;

    // workspace carve-up (256B aligned)
    char* ws = (char*)d_ws;
    size_t offA = (((size_t)N * 4 + 255) / 256) * 256;          // after dinv
    size_t offB = offA + (((size_t)total * 4 + 255) / 256) * 256;
    float* dinv = (float*)ws;
    float* bufA = (float*)(ws + offA);
    float* bufB = (float*)(ws + offB);

    const int B256 = 256;
    dim3 blkN((N + B256 - 1) / B256);
    dim3 blkE((E + B256 - 1) / B256);
    dim3 blkT((total + B256 - 1) / B256);

    const int tilesPerRow = FDIM / 16;                  // 6
    const int nTiles = (N / 16) * tilesPerRow;          // 18750
    dim3 blkG((nTiles * 32 + 127) / 128);               // 4 waves / block

    const int nItems = E + N;
    dim3 blkS(((long long)nItems * 32 + B256 - 1) / B256);

    // --- normalization: deg -> dinv -------------------------------------
    k_fill1<<<blkN, B256, 0, stream>>>(dinv, N);
    k_deg_edges<<<blkE, B256, 0, stream>>>(ei, dinv, E);
    k_rsqrt_inplace<<<blkN, B256, 0, stream>>>(dinv, N);

    // --- layer 1 ---------------------------------------------------------
    k_gemm_wmma_f32<<<blkG, 128, 0, stream>>>(x, W1, bufA, nTiles, tilesPerRow);
    k_zero<<<blkT, B256, 0, stream>>>(bufB, total);
    k_scatter<<<blkS, B256, 0, stream>>>(ei, dinv, bufA, bufB, E, N);
    k_bias_relu<<<blkT, B256, 0, stream>>>(bufB, b1, bufA, total);  // h1 -> bufA

    // --- layer 2 ---------------------------------------------------------
    k_gemm_wmma_f32<<<blkG, 128, 0, stream>>>(bufA, W2, bufB, nTiles, tilesPerRow);
    k_zero<<<blkT, B256, 0, stream>>>(out, total);
    k_scatter<<<blkS, B256, 0, stream>>>(ei, dinv, bufB, out, E, N);
    k_bias_relu<<<blkT, B256, 0, stream>>>(out, b2, out, total);    // in place
}